// EllipsoidRasterizerIntegrandSlang_5222680232018
// MI455X (gfx1250) — compile-verified
//
#include <hip/hip_runtime.h>
#include <math.h>

// ---------------------------------------------------------------------------
// EllipsoidRasterizerIntegrandSlang — CDNA5 (gfx1250) implementation
//
// Pipeline:
//   1) init_kernel               : zero grid region of d_out, counts, header
//   2) transform_kernel [TDM+WMMA]: TENSOR_LOAD_TO_LDS stages centers, then
//                                   v_wmma_f32_16x16x4_f32 computes NDC; emits
//                                   z sort-keys per prim
//   3) 4x radix pass             : stable LSD sort of (zkey, prim)  (1M)
//                                   (wave32 ballot multi-split ranking)
//   4) finalize_kernel           : header[0] = #valid
//   5) gather_kernel      [WMMA] : recompute full 20-float record per sorted
//                                  prim, emit 4 grid (cell,prim) keys
//   6) 3x radix pass             : stable LSD sort of (cell, sortedPrim) (4M)
//   7) grid_first_counts         : first-occurrence index + per-cell counts
//   8) grid_fill / grid_counts   : scatter prim slots + count slot
// ---------------------------------------------------------------------------

typedef __attribute__((ext_vector_type(2))) float    v2f;
typedef __attribute__((ext_vector_type(8))) float    v8f;
typedef __attribute__((ext_vector_type(4))) unsigned u32x4;
typedef __attribute__((ext_vector_type(8))) unsigned u32x8;

namespace er {
constexpr int   N          = 1000000;
constexpr int   G          = 512;
constexpr int   M          = 64;
constexpr float FOCAL      = 768.0f;          // RES/2/(0.5/1.5)
constexpr int   KK         = 2;
constexpr int   REC        = 20;              // floats per result row
constexpr int   GRID_BASE  = 6 + N * REC;     // start of grid region in d_out
constexpr int   GRID_FLOATS= G * G * (M + 1);
constexpr int   TILE       = 4096;            // radix tile (256 thr x 16)
constexpr int   ROUNDS     = 16;
}

// ------------------------------ init ---------------------------------------
__global__ void init_kernel(float* __restrict__ out,
                            const float* __restrict__ background,
                            unsigned* __restrict__ counts,
                            unsigned* __restrict__ validCount) {
  size_t stride = (size_t)gridDim.x * blockDim.x;
  size_t t = (size_t)blockIdx.x * blockDim.x + threadIdx.x;
  for (size_t j = t; j < (size_t)er::GRID_FLOATS; j += stride)
    out[er::GRID_BASE + j] = 0.0f;
  for (size_t j = t; j < (size_t)(er::G * er::G); j += stride)
    counts[j] = 0u;
  if (t == 0) {
    out[1] = (float)er::G;
    out[2] = (float)er::M;
    out[3] = background[0];
    out[4] = background[1];
    out[5] = background[2];
    *validCount = 0u;
  }
}

// --------------------- WMMA helper: 16 points x mvp^T ----------------------
// A (16x4 f32): lane<16 -> {x,y} of point m=lane ; lane>=16 -> {z,1} of m=lane-16
// B (4x16 f32): B[k][n] = mvp[n][k] (n<4), 0 otherwise
// D (16x16)   : lanes 0..3 / 16..19 carry NDC components 0..3 of rows 0..7/8..15
__device__ __forceinline__ void wmma_transform_points(
    const float* __restrict__ sC,   // 256*3 staged centers (block-local)
    const float* __restrict__ mvp,
    float* __restrict__ sNdc,       // 256*4 output (block-local)
    int lane, int wave) {
  v2f b;
  {
    int nn = lane & 15;
    int khalf = (lane >> 4) * 2;                 // K rows 0,1 vs 2,3
    b.x = (nn < 4) ? mvp[nn * 4 + khalf + 0] : 0.0f;
    b.y = (nn < 4) ? mvp[nn * 4 + khalf + 1] : 0.0f;
  }
  int wbase = wave * 32;
  int m = lane & 15;
  int hi = lane >> 4;                            // 0: {x,y}  1: {z,1}
#pragma unroll
  for (int half = 0; half < 2; ++half) {
    int pbase = wbase + half * 16;
    int rowb = (pbase + m) * 3;
    v2f a;
    a.x = sC[rowb + (hi ? 2 : 0)];               // branchless address select
    float ay = sC[rowb + 1];                     // always in-bounds
    a.y = hi ? 1.0f : ay;                        // v_cndmask
    v8f c = {0.f, 0.f, 0.f, 0.f, 0.f, 0.f, 0.f, 0.f};
    v8f d = __builtin_amdgcn_wmma_f32_16x16x4_f32(
        false, a, false, b, (short)0, c, false, false);
    if (m < 4) {
      int rowbase = hi ? 8 : 0;
#pragma unroll
      for (int r = 0; r < 8; ++r)
        sNdc[(pbase + rowbase + r) * 4 + m] = d[r];
    }
  }
}

// --------------------------- transform (keys) ------------------------------
// Wave 0 issues one TENSOR_LOAD_TO_LDS (3 KB contiguous slab of centers) for
// the whole block; tensor_dim0 clamps the tail so OOB elements read as zero.
__global__ void transform_kernel(const float* __restrict__ centers,
                                 const float* __restrict__ mvp,
                                 unsigned* __restrict__ keys,
                                 unsigned* __restrict__ vals,
                                 unsigned* __restrict__ validCount, int n) {
  __shared__ float sC[256 * 3];
  __shared__ float sNdc[256 * 4];
  int tid = threadIdx.x;
  int base = blockIdx.x * 256;
  int i = base + tid;
#if defined(__AMDGCN__)
  if (tid == 0) {
    unsigned long long gaddr =
        (unsigned long long)(const void*)(centers + (size_t)base * 3);
    unsigned ldsOff = (unsigned)(unsigned long long)(void*)&sC[0];
    unsigned rem = (unsigned)((unsigned long long)n * 3 -
                              (unsigned long long)base * 3);  // elements left
    // D# group 0: count=1 | lds_addr | global_addr[56:0] | type=2
    u32x4 g0;
    g0[0] = 1u;
    g0[1] = ldsOff;
    g0[2] = (unsigned)gaddr;
    g0[3] = ((unsigned)(gaddr >> 32) & 0x01FFFFFFu) | (2u << 30);
    // D# group 1: data_size=4B, tensor_dim0=rem, tensor_dim1=1,
    //             tile_dim0=768, tile_dim1=1, tile_dim2=0 (unused),
    //             tensor_dim0_stride=768, tensor_dim1_stride=0
    unsigned td0 = rem, td1 = 1u;
    unsigned tile0 = 768u, tile1 = 1u, tile2 = 0u;
    unsigned long long st0 = 768ull, st1 = 0ull;
    u32x8 g1;
    g1[0] = (2u << 16);                                    // data_size=4B
    g1[1] = (td0 & 0xFFFFu) << 16;                         // dim0[15:0]
    g1[2] = ((td0 >> 16) & 0xFFFFu) | ((td1 & 0xFFFFu) << 16);
    g1[3] = ((td1 >> 16) & 0xFFFFu) | ((tile0 & 0xFFFFu) << 16);
    g1[4] = (tile1 & 0xFFFFu) | ((tile2 & 0xFFFFu) << 16);
    g1[5] = (unsigned)st0;                                 // stride0[31:0]
    g1[6] = ((unsigned)(st0 >> 32) & 0xFFFFu) |
            (((unsigned)st1 & 0xFFFFu) << 16);
    g1[7] = (unsigned)(st1 >> 16);
    asm volatile("tensor_load_to_lds %0, %1" :: "s"(g0), "s"(g1) : "memory");
    __builtin_amdgcn_s_wait_tensorcnt(0);
  }
#else
  if (i < n) {
    sC[tid * 3 + 0] = centers[i * 3 + 0];
    sC[tid * 3 + 1] = centers[i * 3 + 1];
    sC[tid * 3 + 2] = centers[i * 3 + 2];
  }
#endif
  __syncthreads();
  wmma_transform_points(sC, mvp, sNdc, tid & 31, tid >> 5);
  __syncthreads();
  float n2 = sNdc[tid * 4 + 2];
  bool valid = n2 > 0.0f;
  float z = fmaxf(n2, 1e-6f);
  float zmask = valid ? z : 1e30f;               // positive floats: bits monotone
  if (i < n) {
    keys[i] = __float_as_uint(zmask);
    vals[i] = (unsigned)i;
    if (valid) atomicAdd(validCount, 1u);
  }
}

// --------------------------- stable radix sort -----------------------------
__global__ void radix_hist(const unsigned* __restrict__ keys,
                           unsigned* __restrict__ hist, int n, int shift) {
  __shared__ unsigned cnt[256];
  cnt[threadIdx.x] = 0u;
  __syncthreads();
  int base = blockIdx.x * er::TILE;
#pragma unroll 4
  for (int r = 0; r < er::ROUNDS; ++r) {
    int e = base + r * 256 + (int)threadIdx.x;
    if (e < n) {
      unsigned dgt = (keys[e] >> shift) & 255u;
      atomicAdd(&cnt[dgt], 1u);
    }
  }
  __syncthreads();
  hist[blockIdx.x * 256 + threadIdx.x] = cnt[threadIdx.x];
}

__global__ void radix_scan(unsigned* __restrict__ hist,
                           unsigned* __restrict__ digitBase, int tiles) {
  __shared__ unsigned totals[256];
  int d = threadIdx.x;
  unsigned run = 0;
  for (int t = 0; t < tiles; ++t) {
    unsigned v = hist[t * 256 + d];
    hist[t * 256 + d] = run;                     // per-tile exclusive prefix
    run += v;
  }
  totals[d] = run;
  __syncthreads();
  if (d == 0) {
    unsigned acc = 0;
    for (int j = 0; j < 256; ++j) { unsigned v = totals[j]; totals[j] = acc; acc += v; }
  }
  __syncthreads();
  digitBase[d] = totals[d];
}

// Stable scatter: wave32 ballot multi-split for in-wave rank, LDS per-wave
// digit counts for the cross-wave prefix. Order = ascending tid per round,
// rounds ascending => globally stable.
__global__ void radix_scatter(const unsigned* __restrict__ keys,
                              const unsigned* __restrict__ vals,
                              unsigned* __restrict__ dkeys,
                              unsigned* __restrict__ dvals,
                              const unsigned* __restrict__ hist,
                              const unsigned* __restrict__ digitBase,
                              int n, int shift) {
  __shared__ unsigned counter[256];
  __shared__ unsigned waveCnt[8 * 256];
  int tid = threadIdx.x;
  int lane = tid & 31;
  int wave = tid >> 5;
  unsigned ltmask = (1u << lane) - 1u;
  counter[tid] = digitBase[tid] + hist[blockIdx.x * 256 + tid];
  __syncthreads();
  int base = blockIdx.x * er::TILE;
  for (int r = 0; r < er::ROUNDS; ++r) {
#pragma unroll
    for (int w = 0; w < 8; ++w) waveCnt[w * 256 + tid] = 0u;
    __syncthreads();
    int e = base + r * 256 + tid;
    __builtin_prefetch(&keys[e + 256], 0, 1);    // global_prefetch_b8
    unsigned key = 0u, val = 0u, dgt = 256u;     // 256 = out-of-range sentinel
    if (e < n) { key = keys[e]; val = vals[e]; dgt = (key >> shift) & 255u; }
    // peers = lanes in this wave holding the same 9-bit digit value
    unsigned peers = 0xFFFFFFFFu;
#pragma unroll
    for (int bpos = 0; bpos < 9; ++bpos) {
      unsigned bmask = (unsigned)__ballot((dgt >> bpos) & 1u);
      peers &= ((dgt >> bpos) & 1u) ? bmask : ~bmask;
    }
    unsigned rankInWave = (unsigned)__popc(peers & ltmask);
    unsigned cntInWave  = (unsigned)__popc(peers);
    if (dgt < 256u && rankInWave == 0u) waveCnt[wave * 256 + dgt] = cntInWave;
    __syncthreads();
    if (dgt < 256u) {
      unsigned rank = rankInWave;
      for (int w = 0; w < wave; ++w) rank += waveCnt[w * 256 + dgt];
      unsigned pos = counter[dgt] + rank;
      dkeys[pos] = key;
      dvals[pos] = val;
    }
    __syncthreads();
    {
      unsigned total = 0u;
#pragma unroll
      for (int w = 0; w < 8; ++w) total += waveCnt[w * 256 + tid];
      counter[tid] += total;
    }
    __syncthreads();
  }
}

// ------------------------------ finalize -----------------------------------
__global__ void finalize_kernel(float* __restrict__ out,
                                const unsigned* __restrict__ validCount) {
  if (threadIdx.x == 0 && blockIdx.x == 0) out[0] = (float)(*validCount);
}

// ------------------------- gather: records + grid keys ---------------------
__global__ void gather_kernel(const float* __restrict__ centers,
                              const float* __restrict__ scales,
                              const float* __restrict__ rots,
                              const float* __restrict__ colors,
                              const float* __restrict__ opac,
                              const float* __restrict__ mvp,
                              const unsigned* __restrict__ order,
                              float* __restrict__ outResult,   // d_out + 6
                              unsigned* __restrict__ gkeys,
                              unsigned* __restrict__ gvals, int n) {
  __shared__ float sC[256 * 3];
  __shared__ float sNdc[256 * 4];
  int tid = threadIdx.x;
  int s = blockIdx.x * 256 + tid;
  unsigned idx = 0u;
  float px = 0.f, py = 0.f, pz = 0.f;
  if (s < n) {
    idx = order[s];
    px = centers[idx*3+0]; py = centers[idx*3+1]; pz = centers[idx*3+2];
  }
  sC[tid*3+0] = px; sC[tid*3+1] = py; sC[tid*3+2] = pz;
  __syncthreads();
  wmma_transform_points(sC, mvp, sNdc, tid & 31, tid >> 5);
  __syncthreads();
  if (s >= n) return;

  float n0 = sNdc[tid*4+0], n1 = sNdc[tid*4+1], n2 = sNdc[tid*4+2];
  bool valid = n2 > 0.0f;
  float cxs = (n0 + 1.0f) * 0.5f;
  float cys = (n1 + 1.0f) * 0.5f;
  float z = fmaxf(n2, 1e-6f);

  // quaternion -> rotation (reference component order: w,x,y,z)
  float qw = rots[idx*4+0], qx = rots[idx*4+1], qy = rots[idx*4+2], qz = rots[idx*4+3];
  float nrm = fmaxf(sqrtf(qw*qw + qx*qx + qy*qy + qz*qz), 1e-12f);
  float inv = 1.0f / nrm;
  qw *= inv; qx *= inv; qy *= inv; qz *= inv;
  float R00 = 1.f - 2.f*(qy*qy + qz*qz), R01 = 2.f*(qx*qy - qz*qw), R02 = 2.f*(qx*qz + qy*qw);
  float R10 = 2.f*(qx*qy + qz*qw), R11 = 1.f - 2.f*(qx*qx + qz*qz), R12 = 2.f*(qy*qz - qx*qw);
  float R20 = 2.f*(qx*qz - qy*qw), R21 = 2.f*(qy*qz + qx*qw), R22 = 1.f - 2.f*(qx*qx + qy*qy);

  float s0 = fmaxf(scales[idx*3+0], 1e-9f);
  float s1 = fmaxf(scales[idx*3+1], 1e-9f);
  float s2v = fmaxf(scales[idx*3+2], 1e-9f);
  float a0 = s0*s0, a1 = s1*s1, a2 = s2v*s2v;
  float C00 = R00*R00*a0 + R01*R01*a1 + R02*R02*a2;
  float C01 = R00*R10*a0 + R01*R11*a1 + R02*R12*a2;
  float C02 = R00*R20*a0 + R01*R21*a1 + R02*R22*a2;
  float C11 = R10*R10*a0 + R11*R11*a1 + R12*R12*a2;
  float C12 = R10*R20*a0 + R11*R21*a1 + R12*R22*a2;
  float C22 = R20*R20*a0 + R21*R21*a1 + R22*R22*a2;

  float V00 = mvp[0], V01 = mvp[1],  V02 = mvp[2];
  float V10 = mvp[4], V11 = mvp[5],  V12 = mvp[6];
  float T00 = V00*C00 + V01*C01 + V02*C02;
  float T01 = V00*C01 + V01*C11 + V02*C12;
  float T02 = V00*C02 + V01*C12 + V02*C22;
  float T10 = V10*C00 + V11*C01 + V12*C02;
  float T11 = V10*C01 + V11*C11 + V12*C12;
  float T12 = V10*C02 + V11*C12 + V12*C22;
  float cc00 = T00*V00 + T01*V01 + T02*V02;
  float cc01 = T00*V10 + T01*V11 + T02*V12;
  float cc11 = T10*V10 + T11*V11 + T12*V12;

  float sc = er::FOCAL / z;
  float scsq = sc * sc;
  float c00 = cc00 * scsq, c01 = cc01 * scsq, c11 = cc11 * scsq;

  float colr = colors[idx*3+0], colg = colors[idx*3+1], colb = colors[idx*3+2];
  float op = fminf(fmaxf(opac[idx], 0.0f), 1.0f);

  float* r = outResult + (size_t)s * er::REC;
  r[0]  = valid ? 5.0f : 0.0f;
  r[1]  = valid ? cxs : 0.0f;
  r[2]  = valid ? cys : 0.0f;
  r[3]  = valid ? c00 : 0.0f;
  r[4]  = valid ? c01 : 0.0f;
  r[5]  = valid ? c11 : 0.0f;
  r[6]  = 0.0f; r[7] = 0.0f; r[8] = 0.0f;
  r[9]  = valid ? colr : 0.0f;
  r[10] = valid ? colg : 0.0f;
  r[11] = valid ? colb : 0.0f;
  r[12] = valid ? op : 0.0f;
  r[13] = 0.0f; r[14] = 0.0f; r[15] = 0.0f;
  r[16] = 0.0f; r[17] = 0.0f; r[18] = 0.0f; r[19] = 0.0f;

  // grid keys
  int x0 = 0, x1 = -1, y0 = 0, y1 = -1;
  if (valid) {
    float rx = 3.0f * sqrtf(fmaxf(c00, 0.0f)) * (1.0f / (float)er::G);
    float ry = 3.0f * sqrtf(fmaxf(c11, 0.0f)) * (1.0f / (float)er::G);
    float fx0 = floorf((cxs - rx) * (float)er::G);
    float fx1 = floorf((cxs + rx) * (float)er::G);
    float fy0 = floorf((cys - ry) * (float)er::G);
    float fy1 = floorf((cys + ry) * (float)er::G);
    fx0 = fminf(fmaxf(fx0, -2.0e9f), 2.0e9f);
    fx1 = fminf(fmaxf(fx1, -2.0e9f), 2.0e9f);
    fy0 = fminf(fmaxf(fy0, -2.0e9f), 2.0e9f);
    fy1 = fminf(fmaxf(fy1, -2.0e9f), 2.0e9f);
    x0 = (int)fx0; x1 = (int)fx1; y0 = (int)fy0; y1 = (int)fy1;
  }
  const unsigned sentinel = (unsigned)(er::G * er::G);
#pragma unroll
  for (int ky = 0; ky < er::KK; ++ky) {
    int cyc = y0 + ky;
    bool vy = valid && (cyc <= y1) && (cyc >= 0) && (cyc < er::G);
#pragma unroll
    for (int kx = 0; kx < er::KK; ++kx) {
      int cxc = x0 + kx;
      bool vx = (cxc <= x1) && (cxc >= 0) && (cxc < er::G);
      unsigned key = (vy && vx) ? (unsigned)(cyc * er::G + cxc) : sentinel;
      size_t j = (size_t)s * 4 + (size_t)(ky * er::KK + kx);
      gkeys[j] = key;
      gvals[j] = (unsigned)s;
    }
  }
}

// ------------------------------ grid phase ---------------------------------
__global__ void grid_first_counts(const unsigned* __restrict__ sk,
                                  unsigned* __restrict__ first,
                                  unsigned* __restrict__ counts, int n) {
  int i = blockIdx.x * 256 + (int)threadIdx.x;
  if (i >= n) return;
  unsigned k = sk[i];
  if (i == 0 || sk[i - 1] != k) first[k] = (unsigned)i;
  if (k < (unsigned)(er::G * er::G)) atomicAdd(&counts[k], 1u);
}

__global__ void grid_fill(const unsigned* __restrict__ sk,
                          const unsigned* __restrict__ sp,
                          const unsigned* __restrict__ first,
                          float* __restrict__ gridOut, int n) {
  int i = blockIdx.x * 256 + (int)threadIdx.x;
  if (i >= n) return;
  unsigned k = sk[i];
  if (k >= (unsigned)(er::G * er::G)) return;
  unsigned rank = (unsigned)i - first[k];
  if (rank < (unsigned)er::M)
    gridOut[(size_t)k * (er::M + 1) + 1 + rank] = (float)sp[i];
}

__global__ void grid_counts_fill(const unsigned* __restrict__ counts,
                                 float* __restrict__ gridOut) {
  int c = blockIdx.x * 256 + (int)threadIdx.x;
  if (c >= er::G * er::G) return;
  unsigned v = counts[c];
  if (v > (unsigned)er::M) v = (unsigned)er::M;
  gridOut[(size_t)c * (er::M + 1)] = (float)v;
}

// ------------------------------- launcher ----------------------------------
static void run_radix_pass(const unsigned* srcK, const unsigned* srcV,
                           unsigned* dstK, unsigned* dstV,
                           unsigned* hist, unsigned* digitBase,
                           int n, int tiles, int shift, hipStream_t stream) {
  radix_hist<<<tiles, 256, 0, stream>>>(srcK, hist, n, shift);
  radix_scan<<<1, 256, 0, stream>>>(hist, digitBase, tiles);
  radix_scatter<<<tiles, 256, 0, stream>>>(srcK, srcV, dstK, dstV, hist,
                                           digitBase, n, shift);
}

extern "C" void kernel_launch(void* const* d_in, const int* in_sizes, int n_in,
                              void* d_out, int out_size, void* d_ws, size_t ws_size,
                              hipStream_t stream) {
  (void)in_sizes; (void)n_in; (void)out_size; (void)ws_size;
  const float* centers    = (const float*)d_in[0];
  const float* scales     = (const float*)d_in[1];
  const float* rotations  = (const float*)d_in[2];
  const float* colors     = (const float*)d_in[3];
  const float* opacities  = (const float*)d_in[4];
  const float* mvp        = (const float*)d_in[5];
  const float* background = (const float*)d_in[6];
  float* out = (float*)d_out;

  const int n1 = er::N;
  const int n2 = er::N * 4;
  const int tiles1 = (n1 + er::TILE - 1) / er::TILE;   // 245
  const int tiles2 = (n2 + er::TILE - 1) / er::TILE;   // 977

  size_t off = 0;
  auto take = [&](size_t bytes) -> void* {
    void* p = (char*)d_ws + off;
    off += (bytes + 255) & ~(size_t)255;
    return p;
  };
  unsigned* validCount = (unsigned*)take(4);
  unsigned* keysA = (unsigned*)take((size_t)n1 * 4);
  unsigned* keysB = (unsigned*)take((size_t)n1 * 4);
  unsigned* valsA = (unsigned*)take((size_t)n1 * 4);
  unsigned* valsB = (unsigned*)take((size_t)n1 * 4);
  unsigned* hist1 = (unsigned*)take((size_t)tiles1 * 256 * 4);
  unsigned* digitBase = (unsigned*)take(256 * 4);
  unsigned* gk0 = (unsigned*)take((size_t)n2 * 4);
  unsigned* gk1 = (unsigned*)take((size_t)n2 * 4);
  unsigned* gv0 = (unsigned*)take((size_t)n2 * 4);
  unsigned* gv1 = (unsigned*)take((size_t)n2 * 4);
  unsigned* hist2 = (unsigned*)take((size_t)tiles2 * 256 * 4);
  unsigned* first = (unsigned*)take((size_t)(er::G * er::G + 1) * 4);
  unsigned* counts = (unsigned*)take((size_t)(er::G * er::G) * 4);

  // 1) init output / scratch
  init_kernel<<<4096, 256, 0, stream>>>(out, background, counts, validCount);

  // 2) z keys via TDM + WMMA transform
  int pblocks = (n1 + 255) / 256;
  transform_kernel<<<pblocks, 256, 0, stream>>>(centers, mvp, keysA, valsA,
                                                validCount, n1);

  // 3) stable 32-bit radix sort of (zkey, prim): A->B->A->B->A
  run_radix_pass(keysA, valsA, keysB, valsB, hist1, digitBase, n1, tiles1, 0,  stream);
  run_radix_pass(keysB, valsB, keysA, valsA, hist1, digitBase, n1, tiles1, 8,  stream);
  run_radix_pass(keysA, valsA, keysB, valsB, hist1, digitBase, n1, tiles1, 16, stream);
  run_radix_pass(keysB, valsB, keysA, valsA, hist1, digitBase, n1, tiles1, 24, stream);

  // 4) header[0]
  finalize_kernel<<<1, 1, 0, stream>>>(out, validCount);

  // 5) records + grid keys (WMMA again, gathered by sorted order in valsA)
  gather_kernel<<<pblocks, 256, 0, stream>>>(centers, scales, rotations, colors,
                                             opacities, mvp, valsA, out + 6,
                                             gk0, gv0, n1);

  // 6) stable radix sort of (cell, sortedPrim), keys < 2^19: 3 passes, ends in gk1/gv1
  run_radix_pass(gk0, gv0, gk1, gv1, hist2, digitBase, n2, tiles2, 0,  stream);
  run_radix_pass(gk1, gv1, gk0, gv0, hist2, digitBase, n2, tiles2, 8,  stream);
  run_radix_pass(gk0, gv0, gk1, gv1, hist2, digitBase, n2, tiles2, 16, stream);

  // 7) + 8) grid assembly
  int gblocks = (n2 + 255) / 256;
  grid_first_counts<<<gblocks, 256, 0, stream>>>(gk1, first, counts, n2);
  grid_fill<<<gblocks, 256, 0, stream>>>(gk1, gv1, first, out + er::GRID_BASE, n2);
  grid_counts_fill<<<(er::G * er::G + 255) / 256, 256, 0, stream>>>(counts, out + er::GRID_BASE);
}